// DiagonalTransfer_25469156065417
// MI455X (gfx1250) — compile-verified
//
#include <hip/hip_runtime.h>
#include <math.h>

// ---------------------------------------------------------------------------
// out[i,m] = S[m] + log1p(expm1(diag[i]) * exp(xx[i,m] - S[m]))
// S[m]     = logsumexp over rows j of xx[j,m]
// N = M = 4096 fp32.  Pure bandwidth problem (~128MB HBM traffic); xx fits in
// the 192MB L2 so passes 2..4 re-read it from L2, not HBM.
// ---------------------------------------------------------------------------

typedef __attribute__((ext_vector_type(2))) float v2f;
typedef __attribute__((ext_vector_type(8))) float v8f;

#define NCHA 16   // row chunks for the column-max pass
#define NCHB 16   // row chunks for the column-sumexp pass

#if defined(__gfx1250__) && __has_builtin(__builtin_amdgcn_wmma_f32_16x16x4_f32)
#define USE_WMMA_F32 1
#else
#define USE_WMMA_F32 0
#endif

// ---------------------------------------------------------------------------
// Pass 1: partial column max.  Each thread owns 4 columns (float4), walks a
// chunk of rows.  Fully coalesced 512B/wave per iteration.  grid=(M/1024,NCHA)
// ---------------------------------------------------------------------------
__global__ void col_max_partial(const float* __restrict__ xx,
                                float* __restrict__ wmax,
                                int N, int M) {
  const int m4   = blockIdx.x * blockDim.x + threadIdx.x;   // float4 col index
  const int rows = N / NCHA;
  const int r0   = blockIdx.y * rows;
  const int m4s  = M >> 2;                                  // row stride in float4
  const float4* p = (const float4*)xx + (size_t)r0 * m4s + m4;
  float4 mx = make_float4(-3.402823466e38f, -3.402823466e38f,
                          -3.402823466e38f, -3.402823466e38f);
  for (int r = 0; r < rows; ++r) {
    float4 v = *p;
    p += m4s;
    mx.x = fmaxf(mx.x, v.x);
    mx.y = fmaxf(mx.y, v.y);
    mx.z = fmaxf(mx.z, v.z);
    mx.w = fmaxf(mx.w, v.w);
  }
  ((float4*)(wmax + (size_t)blockIdx.y * M))[m4] = mx;
}

// ---------------------------------------------------------------------------
// Pass 2: partial column sum of exp(x - colmax).
// Each wave owns 16 columns.  WMMA trick: with A = all-ones,
//   C[i,n] += sum_k B[k,n]     (V_WMMA_F32_16X16X4_F32, exact f32)
// i.e. one wmma performs an exact column-sum of a 4x16 tile.  Since A is all
// ones the result is invariant to the K<->(vgpr,lane-half) permutation of the
// B layout, so only the N = lane%16 mapping must be honored.
// lanes 0-15  load rows r, r+1   for column cb+lane
// lanes 16-31 load rows r+2, r+3 for column cb+lane-16
// grid = (M/128, NCHB), block = 256 (8 waves).  EXEC is all-ones throughout
// the wmma loop (no divergence until the final masked store).
// ---------------------------------------------------------------------------
__global__ void col_sumexp_partial(const float* __restrict__ xx,
                                   const float* __restrict__ wmax,
                                   float* __restrict__ wsum,
                                   int N, int M) {
  const int lane = threadIdx.x & 31;       // wave32 on gfx1250
  const int wid  = threadIdx.x >> 5;
  const int cb   = blockIdx.x * 128 + wid * 16;
  const int c    = cb + (lane & 15);
  const int half = lane >> 4;

  // combine the NCHA partial maxes for this column (tiny, L2-resident)
  float cm = -3.402823466e38f;
  for (int k = 0; k < NCHA; ++k)
    cm = fmaxf(cm, wmax[(size_t)k * M + c]);

  const int rows = N / NCHB;
  const int r0   = blockIdx.y * rows;
  const float* p0 = xx + (size_t)(r0 + half * 2) * M + c;

#if USE_WMMA_F32
  v8f acc = {};                       // 16x16 f32 accumulator, starts at 0
  const v2f ones = {1.0f, 1.0f};      // A matrix: 16x4 of ones
  for (int rr = 0; rr < rows; rr += 4) {
    __builtin_prefetch(p0 + (size_t)8 * M, 0, 1);   // global_prefetch_b8
    v2f ex;
    ex.x = __expf(p0[0] - cm);
    ex.y = __expf(p0[M] - cm);
    acc = __builtin_amdgcn_wmma_f32_16x16x4_f32(
        /*neg_a=*/false, ones, /*neg_b=*/false, ex,
        /*c_mod=*/(short)0, acc, /*reuse_a=*/false, /*reuse_b=*/false);
    p0 += (size_t)4 * M;
  }
  // every element of C equals colsum[N = lane%16]; take VGPR0
  float colsum = acc[0];
#else
  float accs = 0.0f;
  for (int rr = 0; rr < rows; rr += 4) {
    __builtin_prefetch(p0 + (size_t)8 * M, 0, 1);
    accs += __expf(p0[0] - cm);
    accs += __expf(p0[M] - cm);
    p0 += (size_t)4 * M;
  }
  // lanes L and L^16 hold the two K-halves of the same column
  accs += __shfl_xor(accs, 16, 32);
  float colsum = accs;
#endif

  if (lane < 16)
    wsum[(size_t)blockIdx.y * M + c] = colsum;
}

// ---------------------------------------------------------------------------
// Pass 3: combine partials -> S[m] = colmax[m] + log(sum partials),
// and precompute e1[j] = expm1(diag[j]) once (accurate libm here; tiny kernel)
// ---------------------------------------------------------------------------
__global__ void combine_kernel(const float* __restrict__ wmax,
                               const float* __restrict__ wsum,
                               const float* __restrict__ diag,
                               float* __restrict__ Sb,
                               float* __restrict__ e1,
                               int N, int M) {
  const int t = blockIdx.x * blockDim.x + threadIdx.x;
  if (t < M) {
    float cm = -3.402823466e38f;
    float s  = 0.0f;
    for (int k = 0; k < NCHA; ++k) cm = fmaxf(cm, wmax[(size_t)k * M + t]);
    for (int k = 0; k < NCHB; ++k) s += wsum[(size_t)k * M + t];
    Sb[t] = cm + logf(s);
  }
  if (t < N) e1[t] = expm1f(diag[t]);
}

// ---------------------------------------------------------------------------
// Pass 4: streaming elementwise finalize (L2-resident read of xx, HBM write).
// float4 vectorized; hardware v_exp_f32 / v_log_f32 via __expf/__logf keep the
// transcendental cost below the memory time.  grid=(M/1024, N), block=256.
// ---------------------------------------------------------------------------
__global__ void finalize_kernel(const float* __restrict__ xx,
                                const float* __restrict__ Sb,
                                const float* __restrict__ e1,
                                float* __restrict__ out,
                                int N, int M) {
  const int mb = blockIdx.x * blockDim.x + threadIdx.x;   // float4 col index
  const int j  = blockIdx.y;
  const size_t base = (size_t)j * (M >> 2) + mb;
  float4 x = ((const float4*)xx)[base];
  float4 s = ((const float4*)Sb)[mb];
  const float ej = e1[j];
  float4 o;
  o.x = s.x + __logf(fmaf(ej, __expf(x.x - s.x), 1.0f));
  o.y = s.y + __logf(fmaf(ej, __expf(x.y - s.y), 1.0f));
  o.z = s.z + __logf(fmaf(ej, __expf(x.z - s.z), 1.0f));
  o.w = s.w + __logf(fmaf(ej, __expf(x.w - s.w), 1.0f));
  ((float4*)out)[base] = o;
}

// ---------------------------------------------------------------------------
extern "C" void kernel_launch(void* const* d_in, const int* in_sizes, int n_in,
                              void* d_out, int out_size, void* d_ws, size_t ws_size,
                              hipStream_t stream) {
  const float* xx   = (const float*)d_in[0];
  const float* diag = (const float*)d_in[1];
  float* out = (float*)d_out;

  const int N = in_sizes[1];
  const int M = in_sizes[0] / N;   // 4096 x 4096

  // workspace layout (floats): [NCHA*M max partials][NCHB*M sum partials][M S][N e1]
  float* wmax = (float*)d_ws;
  float* wsum = wmax + (size_t)NCHA * M;
  float* Sb   = wsum + (size_t)NCHB * M;
  float* e1   = Sb + M;

  // Pass 1: partial column max
  dim3 bA(256), gA(M / 1024, NCHA);
  col_max_partial<<<gA, bA, 0, stream>>>(xx, wmax, N, M);

  // Pass 2: partial column sum-of-exp (WMMA column-sum)
  dim3 bB(256), gB(M / 128, NCHB);
  col_sumexp_partial<<<gB, bB, 0, stream>>>(xx, wmax, wsum, N, M);

  // Pass 3: combine -> S, e1
  const int mx = (M > N) ? M : N;
  combine_kernel<<<(mx + 255) / 256, 256, 0, stream>>>(wmax, wsum, diag, Sb, e1, N, M);

  // Pass 4: finalize output
  dim3 bD(256), gD(M / 1024, N);
  finalize_kernel<<<gD, bD, 0, stream>>>(xx, Sb, e1, out, N, M);
}